// SModule_65824668778527
// MI455X (gfx1250) — compile-verified
//
#include <hip/hip_runtime.h>

// Problem constants (from reference)
#define D   512
#define R   64
#define WIN 32
#define BB  4
#define SS  2048
#define NN  (SS + 1)              // 2049 nodes per batch
#define MM  (BB * NN)             // 8196 total rows
#define TILES ((NN + 15) / 16)    // 129 node-tiles per batch
#define WTOT 66                   // 65 window edges + anchor
#define KB  84                    // band K: 80 window rows + anchor + pad to mult-of-4

typedef float v2f __attribute__((ext_vector_type(2)));
typedef float v8f __attribute__((ext_vector_type(8)));

__device__ __forceinline__ int clampi(int x, int lo, int hi) {
    return min(max(x, lo), hi);
}

// ---------------------------------------------------------------------------
// Kernel 1: token embed + pos + input LayerNorm + scalar state projection.
// One block per (b, node). Block n==0 writes the anchor row.
// ---------------------------------------------------------------------------
__global__ __launch_bounds__(256) void embed_ln_kernel(
    const int* __restrict__ ids, const float* __restrict__ emb,
    const float* __restrict__ pos, const float* __restrict__ anchor_val,
    const float* __restrict__ anchor_state, const float* __restrict__ gamma,
    const float* __restrict__ beta, const float* __restrict__ sp_w,
    const float* __restrict__ sp_b, float* __restrict__ val,
    float* __restrict__ state)
{
    __shared__ float sred[256];
    const int bn  = blockIdx.x;
    const int b   = bn / NN;
    const int n   = bn % NN;
    const int tid = threadIdx.x;
    float* vrow = val + (size_t)bn * D;

    if (n == 0) {   // anchor node
        for (int d = tid; d < D; d += 256) vrow[d] = anchor_val[d];
        if (tid == 0) state[bn] = anchor_state[0];
        return;
    }
    const int s  = n - 1;
    const int id = ids[b * SS + s];
    const float* erow = emb + (size_t)id * D;
    const float* prow = pos + (size_t)s * D;
    const int d0 = tid, d1 = tid + 256;
    float x0 = erow[d0] + prow[d0];
    float x1 = erow[d1] + prow[d1];

    sred[tid] = x0 + x1; __syncthreads();
    for (int o = 128; o > 0; o >>= 1) { if (tid < o) sred[tid] += sred[tid + o]; __syncthreads(); }
    float mean = sred[0] * (1.0f / D);
    __syncthreads();
    float c0 = x0 - mean, c1 = x1 - mean;
    sred[tid] = c0 * c0 + c1 * c1; __syncthreads();
    for (int o = 128; o > 0; o >>= 1) { if (tid < o) sred[tid] += sred[tid + o]; __syncthreads(); }
    float rstd = rsqrtf(sred[0] * (1.0f / D) + 1e-5f);
    __syncthreads();
    float y0 = c0 * rstd * gamma[d0] + beta[d0];
    float y1 = c1 * rstd * gamma[d1] + beta[d1];
    vrow[d0] = y0; vrow[d1] = y1;
    sred[tid] = y0 * sp_w[d0] + y1 * sp_w[d1]; __syncthreads();
    for (int o = 128; o > 0; o >>= 1) { if (tid < o) sred[tid] += sred[tid + o]; __syncthreads(); }
    if (tid == 0) state[bn] = sred[0] + sp_b[0];
}

// ---------------------------------------------------------------------------
// Kernel 2: q = val @ wq, k = val @ wk  via V_WMMA_F32_16X16X4_F32.
// One wave per 16-row M tile; 4 N tiles (R=64) per matrix; q and k share the
// A fragment each k-step. Fragment layouts per CDNA5 ISA 7.12.2.
// ---------------------------------------------------------------------------
__global__ __launch_bounds__(256) void qk_gemm_kernel(
    const float* __restrict__ val, const float* __restrict__ wq,
    const float* __restrict__ wk, float* __restrict__ q,
    float* __restrict__ k)
{
    const int lane   = threadIdx.x & 31;
    const int wave   = threadIdx.x >> 5;
    const int mtile  = blockIdx.x * 8 + wave;
    const int mtiles = (MM + 15) / 16;
    if (mtile >= mtiles) return;

    const int row0 = mtile * 16;
    const int lm   = lane & 15;
    const int lh   = lane >> 4;
    const int arow = min(row0 + lm, MM - 1);        // clamp reads on ragged tile
    const float* aptr = val + (size_t)arow * D + 2 * lh;

    v8f accq[4] = {};
    v8f acck[4] = {};

    for (int kk = 0; kk < D; kk += 4) {
        v2f a = *(const v2f*)(aptr + kk);
        const float* bqp = wq + (size_t)(kk + 2 * lh) * R + lm;
        const float* bkp = wk + (size_t)(kk + 2 * lh) * R + lm;
#pragma unroll
        for (int nt = 0; nt < 4; ++nt) {
            v2f bq; bq.x = bqp[nt * 16]; bq.y = bqp[R + nt * 16];
            v2f bk; bk.x = bkp[nt * 16]; bk.y = bkp[R + nt * 16];
            accq[nt] = __builtin_amdgcn_wmma_f32_16x16x4_f32(
                false, a, false, bq, (short)0, accq[nt], false, false);
            acck[nt] = __builtin_amdgcn_wmma_f32_16x16x4_f32(
                false, a, false, bk, (short)0, acck[nt], false, false);
        }
    }

#pragma unroll
    for (int nt = 0; nt < 4; ++nt) {
#pragma unroll
        for (int i = 0; i < 8; ++i) {
            const int r = row0 + i + 8 * lh;
            if (r < MM) {
                q[(size_t)r * R + nt * 16 + lm] = accq[nt][i];
                k[(size_t)r * R + nt * 16 + lm] = acck[nt][i];
            }
        }
    }
}

// ---------------------------------------------------------------------------
// Kernel 3 (fused, WMMA-based): one block = 16 consecutive nodes of a batch.
//  Phase 1: scores[16x80] = Q_tile[16x64] . K_band^T[64x80]   (5 waves, WMMA)
//  Phase 1.5: signed-|.|-softmax per node -> band weights W[16x84] in LDS,
//             plus dstate (16 threads, fixed-order = deterministic)
//  Phase 2: dval[16x512] = W[16x84] . V_band[84x512]          (8 waves, WMMA)
//           band row k<80 -> val row p0-32+k (clamped; W==0 where invalid),
//           k==80 -> anchor row, k=81..83 -> zero padding.
//  Phase 3: residual + LayerNorm from LDS y-tile, deterministic reductions.
// ---------------------------------------------------------------------------
__global__ __launch_bounds__(256) void attn_fused_kernel(
    const float* __restrict__ val_in, const float* __restrict__ state_in,
    const float* __restrict__ q, const float* __restrict__ k,
    const float* __restrict__ gamma, const float* __restrict__ beta,
    float* __restrict__ val_out, float* __restrict__ state_out)
{
    __shared__ float s_scores[16][80];
    __shared__ float s_w[16][KB];
    __shared__ float s_y[16][D];
    __shared__ float s_r[16][16];
    __shared__ float s_mean[16];
    __shared__ float s_rstd[16];

    const int tid  = threadIdx.x;
    const int lane = tid & 31;
    const int wave = tid >> 5;
    const int lm   = lane & 15;
    const int lh   = lane >> 4;
    const int b    = blockIdx.x / TILES;
    const int t    = blockIdx.x % TILES;
    const int p0   = t * 16;
    const int base = b * NN;

    // ---- Phase 1: window scores via WMMA -------------------------------
    if (wave < 5) {
        const int qrow = base + clampi(p0 + lm, 0, NN - 1);
        const int j    = wave * 16 + lm;                     // band column
        const int krow = base + clampi(p0 - 32 + j, 0, NN - 1);
        const float* qp = q + (size_t)qrow * R + 2 * lh;
        const float* kp = k + (size_t)krow * R + 2 * lh;
        v8f acc = {};
        for (int kk = 0; kk < R; kk += 4) {
            v2f a  = *(const v2f*)(qp + kk);
            v2f bb = *(const v2f*)(kp + kk);
            acc = __builtin_amdgcn_wmma_f32_16x16x4_f32(
                false, a, false, bb, (short)0, acc, false, false);
        }
#pragma unroll
        for (int i = 0; i < 8; ++i)
            s_scores[i + 8 * lh][wave * 16 + lm] = acc[i];
    }
    __syncthreads();

    // zero the band weight matrix
    for (int x = tid; x < 16 * KB; x += 256) (&s_w[0][0])[x] = 0.f;
    __syncthreads();

    // ---- Phase 1.5: softmax + band weights + dstate --------------------
    if (tid < 16) {
        const int m    = tid;
        const int node = p0 + m;
        if (node < NN) {
            const float scale = 0.125f;            // 1/sqrt(R)
            // anchor score
            float adot = 0.f;
            {
                const float* qp = q + (size_t)(base + node) * R;
                const float* k0 = k + (size_t)base * R;
                for (int r = 0; r < R; ++r) adot = fmaf(qp[r], k0[r], adot);
                adot *= scale;
            }
            const bool avalid = (node - WIN) > 0;
            // pass A: max |score| over valid edges
            float mx = avalid ? fabsf(adot) : -1e30f;
            for (int w = 0; w < 65; ++w) {
                const int p = node - WIN + w;
                if (p >= 0 && p < NN)
                    mx = fmaxf(mx, fabsf(s_scores[m][m + w] * scale));
            }
            // pass B: exp & sum; stash signed exp back into s_scores row m
            float sum = 0.f;
            for (int w = 0; w < 65; ++w) {
                const int p = node - WIN + w;
                float se = 0.f;
                if (p >= 0 && p < NN) {
                    const float s = s_scores[m][m + w] * scale;
                    const float e = __expf(fabsf(s) - mx);
                    sum += e;
                    se = (s > 0.f) ? e : ((s < 0.f) ? -e : 0.f);
                }
                s_scores[m][m + w] = se;
            }
            float ase = 0.f;
            if (avalid) {
                const float e = __expf(fabsf(adot) - mx);
                sum += e;
                ase = (adot > 0.f) ? e : ((adot < 0.f) ? -e : 0.f);
            }
            // pass C: normalize -> band matrix, accumulate dstate
            const float inv = 1.f / sum;
            float ds = 0.f;
            for (int w = 0; w < 65; ++w) {
                const float we = s_scores[m][m + w] * inv;
                s_w[m][m + w] = we;
                ds += we * state_in[base + clampi(node - WIN + w, 0, NN - 1)];
            }
            const float wa = ase * inv;
            s_w[m][80] = wa;
            ds += wa * state_in[base];
            state_out[base + node] = state_in[base + node] + ds;
        }
    }
    __syncthreads();

    // ---- Phase 2: dval GEMM, each wave owns 64 channels ----------------
    {
        const int c0 = wave * 64;
        v8f acc[4] = {};
        for (int kk = 0; kk < KB; kk += 4) {
            const int kp0 = kk + 2 * lh;
            const int kp1 = kp0 + 1;
            v2f a;
            a.x = s_w[lm][kp0];
            a.y = s_w[lm][kp1];
            const int r0 = (kp0 >= 80) ? base : base + clampi(p0 - 32 + kp0, 0, NN - 1);
            const int r1 = (kp1 >= 80) ? base : base + clampi(p0 - 32 + kp1, 0, NN - 1);
            const float* v0 = val_in + (size_t)r0 * D + c0 + lm;
            const float* v1 = val_in + (size_t)r1 * D + c0 + lm;
#pragma unroll
            for (int nt = 0; nt < 4; ++nt) {
                v2f bb;
                bb.x = v0[nt * 16];
                bb.y = v1[nt * 16];
                acc[nt] = __builtin_amdgcn_wmma_f32_16x16x4_f32(
                    false, a, false, bb, (short)0, acc[nt], false, false);
            }
        }
        // residual add + stage y tile into LDS
#pragma unroll
        for (int nt = 0; nt < 4; ++nt) {
#pragma unroll
            for (int i = 0; i < 8; ++i) {
                const int m    = i + 8 * lh;
                const int col  = c0 + nt * 16 + lm;
                const int node = p0 + m;
                float y = 0.f;
                if (node < NN)
                    y = acc[nt][i] + val_in[(size_t)(base + node) * D + col];
                s_y[m][col] = y;
            }
        }
    }
    __syncthreads();

    // ---- Phase 3: per-node LayerNorm (16 threads per node) -------------
    const int m  = tid >> 4;
    const int jj = tid & 15;
    float part = 0.f;
    for (int e = 0; e < 32; ++e) part += s_y[m][jj * 32 + e];
    s_r[m][jj] = part;
    __syncthreads();
    if (jj == 0) {
        float tot = 0.f;
        for (int x = 0; x < 16; ++x) tot += s_r[m][x];
        s_mean[m] = tot * (1.0f / D);
    }
    __syncthreads();
    const float mean = s_mean[m];
    part = 0.f;
    for (int e = 0; e < 32; ++e) {
        const float c = s_y[m][jj * 32 + e] - mean;
        part += c * c;
    }
    s_r[m][jj] = part;
    __syncthreads();
    if (jj == 0) {
        float tot = 0.f;
        for (int x = 0; x < 16; ++x) tot += s_r[m][x];
        s_rstd[m] = rsqrtf(tot * (1.0f / D) + 1e-5f);
    }
    __syncthreads();
    const int node = p0 + m;
    if (node < NN) {
        const float rstd = s_rstd[m];
        float* vo = val_out + (size_t)(base + node) * D;
        for (int e = 0; e < 32; ++e) {
            const int col = jj * 32 + e;
            vo[col] = (s_y[m][col] - mean) * rstd * gamma[col] + beta[col];
        }
    }
}

// ---------------------------------------------------------------------------
// Host launcher
// ---------------------------------------------------------------------------
extern "C" void kernel_launch(void* const* d_in, const int* in_sizes, int n_in,
                              void* d_out, int out_size, void* d_ws, size_t ws_size,
                              hipStream_t stream)
{
    const int*   ids    = (const int*)  d_in[0];
    const float* emb    = (const float*)d_in[1];
    const float* pos    = (const float*)d_in[2];
    const float* aval   = (const float*)d_in[3];
    const float* astate = (const float*)d_in[4];
    const float* ing    = (const float*)d_in[5];
    const float* inb    = (const float*)d_in[6];
    const float* spw    = (const float*)d_in[7];
    const float* spb    = (const float*)d_in[8];
    const float* wq     = (const float*)d_in[9];
    const float* wk     = (const float*)d_in[10];
    const float* lng    = (const float*)d_in[11];
    const float* lnb    = (const float*)d_in[12];

    float* ws = (float*)d_ws;
    size_t o = 0;
    float* val_a = ws + o; o += (size_t)MM * D;
    float* val_b = ws + o; o += (size_t)MM * D;
    float* st_a  = ws + o; o += MM;
    float* st_b  = ws + o; o += MM;
    float* qbuf  = ws + o; o += (size_t)MM * R;
    float* kbuf  = ws + o; o += (size_t)MM * R;

    float* out_state = (float*)d_out;      // [B*N]
    float* out_val   = out_state + MM;     // [B*N*D]

    embed_ln_kernel<<<MM, 256, 0, stream>>>(ids, emb, pos, aval, astate,
                                            ing, inb, spw, spb, val_a, st_a);

    const int mtiles  = (MM + 15) / 16;      // 513
    const int gblocks = (mtiles + 7) / 8;    // 8 waves/block
    const int ablocks = BB * TILES;          // 516

    // layer 0: val_a -> val_b
    qk_gemm_kernel<<<gblocks, 256, 0, stream>>>(val_a, wq, wk, qbuf, kbuf);
    attn_fused_kernel<<<ablocks, 256, 0, stream>>>(val_a, st_a, qbuf, kbuf,
                                                   lng, lnb, val_b, st_b);
    // layer 1: val_b -> d_out
    qk_gemm_kernel<<<gblocks, 256, 0, stream>>>(val_b, wq + (size_t)D * R,
                                                wk + (size_t)D * R, qbuf, kbuf);
    attn_fused_kernel<<<ablocks, 256, 0, stream>>>(val_b, st_b, qbuf, kbuf,
                                                   lng + D, lnb + D, out_val, out_state);
}